// GATLayer_29901562315450
// MI455X (gfx1250) — compile-verified
//
#include <hip/hip_runtime.h>

typedef float v2f __attribute__((ext_vector_type(2)));
typedef float v8f __attribute__((ext_vector_type(8)));

#define IN_DIM 128
#define OUT_DIM 64
#define NEG_SLOPE 0.01f
#define LDS_STRIDE 130  // 128 + 2 pad words: lane stride 130 % 64 banks -> conflict-free-ish

__device__ __forceinline__ void atomic_max_f32(float* addr, float val) {
    // monotonic-bits trick; e_max initialized to -inf
    if (val >= 0.0f) {
        atomicMax((int*)addr, __float_as_int(val));
    } else {
        atomicMin((unsigned int*)addr, __float_as_uint(val));
    }
}

// ---------------------------------------------------------------------------
// Kernel 1: zero output accumulator, init segment-max to -inf, denom to 0
// ---------------------------------------------------------------------------
__global__ void k_init(float* __restrict__ out, float* __restrict__ e_max,
                       float* __restrict__ denom, int n_nodes) {
    int idx = blockIdx.x * blockDim.x + threadIdx.x;
    if (idx < n_nodes * OUT_DIM) out[idx] = 0.0f;
    if (idx < n_nodes) {
        e_max[idx] = -__builtin_inff();
        denom[idx] = 0.0f;
    }
}

// ---------------------------------------------------------------------------
// Kernel 2: z = h @ fc_w^T via V_WMMA_F32_16X16X4_F32, fused attention dots
//   one wave per 16-row tile; 4 C-tiles (16x16) cover OUT_DIM=64
//   fc_w [64x128] staged to LDS (row stride 130 words)
// ---------------------------------------------------------------------------
__global__ void __launch_bounds__(256)
k_gemm_attn(const float* __restrict__ h, const float* __restrict__ fc_w,
            const float* __restrict__ attn_w,
            float* __restrict__ z, float* __restrict__ a_src,
            float* __restrict__ a_dst, int n_nodes) {
    __shared__ float lds_w[OUT_DIM * LDS_STRIDE];

    // cooperative coalesced stage of fc_w
    for (int i = threadIdx.x; i < OUT_DIM * IN_DIM; i += blockDim.x) {
        lds_w[(i >> 7) * LDS_STRIDE + (i & 127)] = fc_w[i];
    }
    __syncthreads();

    const int wave = threadIdx.x >> 5;
    const int lane = threadIdx.x & 31;
    const int tile = blockIdx.x * 8 + wave;
    const int row_base = tile * 16;
    if (row_base >= n_nodes) return;   // wave-uniform: EXEC stays all-ones for WMMA

    const int half = lane >> 4;        // 0: lanes 0-15, 1: lanes 16-31
    const int l16  = lane & 15;
    const int koff = half * 2;         // ISA 16x4 f32 A/B layout: K pairs per half-wave

    v8f acc[4];
    #pragma unroll
    for (int t = 0; t < 4; ++t) acc[t] = (v8f){0,0,0,0,0,0,0,0};

    const float* arow = h + (size_t)(row_base + l16) * IN_DIM;

    for (int k0 = 0; k0 < IN_DIM; k0 += 4) {
        // A fragment: A[m=l16, k0+koff .. k0+koff+1]
        v2f a = *(const v2f*)(arow + k0 + koff);
        #pragma unroll
        for (int t = 0; t < 4; ++t) {
            // B fragment: B[k][n] = fc_w[n][k];  n = t*16 + l16, k = k0+koff{,+1}
            v2f b = *(const v2f*)(&lds_w[(t * 16 + l16) * LDS_STRIDE + k0 + koff]);
            acc[t] = __builtin_amdgcn_wmma_f32_16x16x4_f32(
                false, a, false, b, (short)0, acc[t], false, false);
        }
    }

    // C/D layout: lane -> N = l16 (per half), VGPR r -> M = r + 8*half
    const int mofs = half * 8;
    float ps[8], pd[8];
    #pragma unroll
    for (int r = 0; r < 8; ++r) { ps[r] = 0.0f; pd[r] = 0.0f; }

    #pragma unroll
    for (int t = 0; t < 4; ++t) {
        float wsrc = attn_w[t * 16 + l16];
        float wdst = attn_w[OUT_DIM + t * 16 + l16];
        #pragma unroll
        for (int r = 0; r < 8; ++r) {
            float v = acc[t][r];
            z[(size_t)(row_base + mofs + r) * OUT_DIM + t * 16 + l16] = v;
            ps[r] += v * wsrc;
            pd[r] += v * wdst;
        }
    }
    // reduce over the 16 lanes of each half-wave (N dimension)
    #pragma unroll
    for (int r = 0; r < 8; ++r) {
        #pragma unroll
        for (int s = 1; s < 16; s <<= 1) {
            ps[r] += __shfl_xor(ps[r], s, 16);
            pd[r] += __shfl_xor(pd[r], s, 16);
        }
    }
    if (l16 == 0) {
        #pragma unroll
        for (int r = 0; r < 8; ++r) {
            a_src[row_base + mofs + r] = ps[r];
            a_dst[row_base + mofs + r] = pd[r];
        }
    }
}

// ---------------------------------------------------------------------------
// Kernel 3: per-edge leaky-relu logit + segment max (atomic f32 max)
// ---------------------------------------------------------------------------
__global__ void k_edge_logit(const int* __restrict__ src, const int* __restrict__ dst,
                             const float* __restrict__ a_src, const float* __restrict__ a_dst,
                             float* __restrict__ e_buf, float* __restrict__ e_max, int n_edges) {
    int i = blockIdx.x * blockDim.x + threadIdx.x;
    if (i >= n_edges) return;
    float e = a_src[src[i]] + a_dst[dst[i]];
    e = (e >= 0.0f) ? e : NEG_SLOPE * e;
    e_buf[i] = e;
    atomic_max_f32(&e_max[dst[i]], e);
}

// ---------------------------------------------------------------------------
// Kernel 4: ex = exp(e - e_max[dst]); denom[dst] += ex
// ---------------------------------------------------------------------------
__global__ void k_edge_exp(const int* __restrict__ dst, const float* __restrict__ e_max,
                           float* __restrict__ e_buf, float* __restrict__ denom, int n_edges) {
    int i = blockIdx.x * blockDim.x + threadIdx.x;
    if (i >= n_edges) return;
    int d = dst[i];
    float ex = __expf(e_buf[i] - e_max[d]);
    e_buf[i] = ex;
    atomicAdd(&denom[d], ex);
}

// ---------------------------------------------------------------------------
// Kernel 5: h_out[dst] += (ex/denom[dst]) * z[src]; one wave per edge,
// lanes cover 64 dims (2 per lane). All targets L2-resident (192 MB).
// ---------------------------------------------------------------------------
__global__ void k_scatter(const int* __restrict__ src, const int* __restrict__ dst,
                          const float* __restrict__ e_buf, const float* __restrict__ denom,
                          const float* __restrict__ z, float* __restrict__ out, int n_edges) {
    int gid = blockIdx.x * blockDim.x + threadIdx.x;
    int i = gid >> 5;
    if (i >= n_edges) return;
    int lane = gid & 31;
    int s = src[i];
    int d = dst[i];
    float alpha = e_buf[i] / denom[d];
    const float* zr = z + (size_t)s * OUT_DIM;
    float* orow = out + (size_t)d * OUT_DIM;
    atomicAdd(&orow[lane],      alpha * zr[lane]);
    atomicAdd(&orow[lane + 32], alpha * zr[lane + 32]);
}

// ---------------------------------------------------------------------------
extern "C" void kernel_launch(void* const* d_in, const int* in_sizes, int n_in,
                              void* d_out, int out_size, void* d_ws, size_t ws_size,
                              hipStream_t stream) {
    const float* h      = (const float*)d_in[0];
    const int*   src    = (const int*)d_in[1];
    const int*   dst    = (const int*)d_in[2];
    const float* fc_w   = (const float*)d_in[3];
    const float* attn_w = (const float*)d_in[4];
    float* out = (float*)d_out;

    const int n_nodes = in_sizes[0] / IN_DIM;
    const int n_edges = in_sizes[1];

    // carve workspace
    char* ws = (char*)d_ws;
    float* z      = (float*)ws; ws += (size_t)n_nodes * OUT_DIM * sizeof(float);
    float* a_src  = (float*)ws; ws += (size_t)n_nodes * sizeof(float);
    float* a_dst  = (float*)ws; ws += (size_t)n_nodes * sizeof(float);
    float* e_max  = (float*)ws; ws += (size_t)n_nodes * sizeof(float);
    float* denom  = (float*)ws; ws += (size_t)n_nodes * sizeof(float);
    float* e_buf  = (float*)ws; ws += (size_t)n_edges * sizeof(float);

    const int T = 256;

    k_init<<<(n_nodes * OUT_DIM + T - 1) / T, T, 0, stream>>>(out, e_max, denom, n_nodes);

    int n_tiles = (n_nodes + 15) / 16;   // N = 100000 is a multiple of 16
    k_gemm_attn<<<(n_tiles + 7) / 8, 256, 0, stream>>>(h, fc_w, attn_w, z, a_src, a_dst, n_nodes);

    k_edge_logit<<<(n_edges + T - 1) / T, T, 0, stream>>>(src, dst, a_src, a_dst, e_buf, e_max, n_edges);
    k_edge_exp<<<(n_edges + T - 1) / T, T, 0, stream>>>(dst, e_max, e_buf, denom, n_edges);

    long total = (long)n_edges * 32;
    k_scatter<<<(int)((total + T - 1) / T), T, 0, stream>>>(src, dst, e_buf, denom, z, out, n_edges);
}